// MyLayer_86165633892423
// MI455X (gfx1250) — compile-verified
//
#include <hip/hip_runtime.h>
#include <hip/hip_bf16.h>

typedef __attribute__((ext_vector_type(2))) float v2f;
typedef __attribute__((ext_vector_type(8))) float v8f;

#define MNODES 10000
#define NBATCH 16
#define FIN    32
#define FOUT   64
#define KCH    5
#define NNZV   160000
#define CDIM   512          // FIN * NBATCH
#define MPOOL  2500         // MNODES / 4

// GEMM LDS layout (floats):
//  A panels row-major [5][16][514]: stride 514 dwords -> rows 8B aligned for
//  async B64 chunks; 514%64==2 -> A-frag reads are bank-conflict free
//  (lanes 0-15 banks c0+2l, lanes 16-31 banks c0+32+2l).
//  B staged pre-swizzled in WMMA fragment order: [q][tile][lane][2]
//  -> one aligned ds_load_b64 per fragment, all 64 banks, zero re-pair movs.
#define LDSA_RSTR 514
#define LDSA_K    (16 * LDSA_RSTR)     // 8224
#define LDSA_TOT  (5 * LDSA_K)         // 41120
#define LDSB_TOT  (40 * 4 * 64)        // 10240

__device__ __forceinline__ float4 f4fma(float s, float4 a, float4 acc) {
    acc.x += s * a.x; acc.y += s * a.y; acc.z += s * a.z; acc.w += s * a.w;
    return acc;
}

// ---------------------------------------------------------------------------
// X0[m, f*16 + n] = x[n, m, f]   (column layout c = f*N + n, N == 16)
// LDS tile transpose: coalesced 128B reads per wave AND contiguous 2KB row
// writes. 17-padded tile is bank-conflict free both directions.
// ---------------------------------------------------------------------------
__global__ void __launch_bounds__(512)
reorder_x(const float* __restrict__ x, float* __restrict__ X0) {
    __shared__ float buf[32 * 17];
    const int t = threadIdx.x;
    const int n = t >> 5;              // wave id = batch index
    const int f = t & 31;              // lane = feature
    const int mbase = blockIdx.x * 16;
    for (int mi = 0; mi < 16; ++mi) {
        int m = mbase + mi;
        float v = x[((size_t)n * MNODES + m) * FIN + f];
        __syncthreads();               // protect previous iteration's reads
        buf[f * 17 + n] = v;
        __syncthreads();
        X0[(size_t)m * CDIM + t] = buf[(t >> 4) * 17 + (t & 15)];
    }
}

// ---------------------------------------------------------------------------
// CSR build: zero counts -> count -> single-block scan -> scatter
// ---------------------------------------------------------------------------
__global__ void csr_zero(int* __restrict__ cnt) {
    int i = blockIdx.x * blockDim.x + threadIdx.x;
    if (i < MNODES) cnt[i] = 0;
}

__global__ void csr_count(const int* __restrict__ rows, int* __restrict__ cnt) {
    int e = blockIdx.x * blockDim.x + threadIdx.x;
    if (e < NNZV) atomicAdd(&cnt[rows[e]], 1);
}

__global__ void __launch_bounds__(256)
csr_scan(const int* __restrict__ cnt, int* __restrict__ offs, int* __restrict__ cursor) {
    __shared__ int part[256];
    int t = threadIdx.x;
    int begin = t * 40;               // 256*40 = 10240 >= MNODES
    int end   = begin + 40;
    if (begin > MNODES) begin = MNODES;
    if (end   > MNODES) end   = MNODES;
    int s = 0;
    for (int i = begin; i < end; ++i) s += cnt[i];
    part[t] = s;
    __syncthreads();
    for (int off = 1; off < 256; off <<= 1) {   // Hillis-Steele inclusive scan
        int v = (t >= off) ? part[t - off] : 0;
        __syncthreads();
        part[t] += v;
        __syncthreads();
    }
    int run = part[t] - s;            // exclusive prefix of this chunk
    for (int i = begin; i < end; ++i) {
        offs[i] = run; cursor[i] = run; run += cnt[i];
    }
    if (t == 255) offs[MNODES] = part[255];
}

__global__ void csr_scatter(const int* __restrict__ rows, const int* __restrict__ cols,
                            const float* __restrict__ vals, int* __restrict__ cursor,
                            int* __restrict__ scols, float* __restrict__ svals) {
    int e = blockIdx.x * blockDim.x + threadIdx.x;
    if (e < NNZV) {
        int r = rows[e];
        int p = atomicAdd(&cursor[r], 1);
        scols[p] = cols[e];
        svals[p] = vals[e];
    }
}

// ---------------------------------------------------------------------------
// Fused gather-SpMM + Chebyshev combine:
//   Y[r,:] = sA * sum_e vals[e]*X1[col[e],:]  + a1*X1[r,:] + a0*X0[r,:]
// One wave per row; each lane owns 16 of the 512 columns as 4x float4.
// All gathered rows are L2-resident (X1 is 20MB << 192MB L2).
// ---------------------------------------------------------------------------
__global__ void cheb_spmm_gather(const int* __restrict__ offs, const int* __restrict__ scols,
                                 const float* __restrict__ svals,
                                 const float* __restrict__ X1, const float* __restrict__ X0,
                                 float* __restrict__ Y, float sA, float a1, float a0) {
    int gw   = (blockIdx.x * blockDim.x + threadIdx.x) >> 5;
    int lane = threadIdx.x & 31;
    if (gw >= MNODES) return;
    int e0 = offs[gw], e1 = offs[gw + 1];

    float4 z = {0.f, 0.f, 0.f, 0.f};
    float4 acc0 = z, acc1 = z, acc2 = z, acc3 = z;

    for (int e = e0; e < e1; ++e) {
        int   c = scols[e];
        float v = svals[e];
        const float4* xs = (const float4*)(X1 + (size_t)c * CDIM);
        if (e + 1 < e1)   // hide gather latency: prefetch next neighbor row
            __builtin_prefetch(X1 + (size_t)scols[e + 1] * CDIM, 0, 3);
        acc0 = f4fma(v, xs[lane],      acc0);
        acc1 = f4fma(v, xs[32 + lane], acc1);
        acc2 = f4fma(v, xs[64 + lane], acc2);
        acc3 = f4fma(v, xs[96 + lane], acc3);
    }

    const float4* u4 = (const float4*)(X1 + (size_t)gw * CDIM);
    const float4* w4 = (const float4*)(X0 + (size_t)gw * CDIM);
    float4*       y4 = (float4*)(Y + (size_t)gw * CDIM);
    float4 accs[4] = {acc0, acc1, acc2, acc3};
#pragma unroll
    for (int j = 0; j < 4; ++j) {
        float4 a = accs[j];
        float4 u = u4[j * 32 + lane];
        float4 w = w4[j * 32 + lane];
        float4 r;
        r.x = sA * a.x + a1 * u.x + a0 * w.x;
        r.y = sA * a.y + a1 * u.y + a0 * w.y;
        r.z = sA * a.z + a1 * u.z + a0 * w.z;
        r.w = sA * a.w + a1 * u.w + a0 * w.w;
        y4[j * 32 + lane] = r;
    }
}

// ---------------------------------------------------------------------------
// Fused WMMA GEMM + bias + relu + maxpool(4).
// Workgroup: 512 threads = 16 waves, one 16-node tile. Wave w = batch n.
// A panels staged with CDNA5 async global->LDS B64 copies (ASYNCcnt);
// B staged pre-swizzled so each fragment is one ds_load_b64.
// 40 quads of V_WMMA_F32_16X16X4_F32 x 4 FOUT tiles per wave.
// ---------------------------------------------------------------------------
__global__ void __launch_bounds__(512)
cheb_gemm(const float* __restrict__ Xall, const float* __restrict__ Wk,
          const float* __restrict__ bias, float* __restrict__ out) {
    extern __shared__ float smem[];
    float* ldsA = smem;                // [5][16][514]  ([k][mr][c])
    float* ldsB = smem + LDSA_TOT;     // [40][4][32][2] fragment order

    const int t  = threadIdx.x;
    const int m0 = blockIdx.x * 16;

    // Stage B pre-swizzled: entry i = ((q*4+tile)*32 + lane)*2 + v holds
    // W[rowOf(q*4 + 2*(lane>=16) + v), tile*16 + lane&15],
    // rowOf(flat) = (flat%32)*5 + flat/32  (reduction order flat = k*32+f).
    for (int i = t; i < LDSB_TOT; i += 512) {
        int v    = i & 1;
        int p    = i >> 1;
        int lane = p & 31;
        int tile = (p >> 5) & 3;
        int q    = p >> 7;
        int flat = q * 4 + ((lane >> 4) << 1) + v;
        int row  = (flat & 31) * 5 + (flat >> 5);
        ldsB[i] = Wk[row * FOUT + tile * 16 + (lane & 15)];
    }

    // Async-stage A panels: row m0+mr, column c of buffer k goes to LDS panel k, row mr, column c.
    // 8-byte chunks, 8 per thread per k (4096 chunks / 512 threads).
    const uint32_t ldsA_base = (uint32_t)(uintptr_t)ldsA;
#pragma unroll
    for (int k = 0; k < 5; ++k) {
        const float* Xp = Xall + (size_t)k * MNODES * CDIM + (size_t)m0 * CDIM;
#pragma unroll
        for (int pass = 0; pass < 8; ++pass) {
            int id = pass * 512 + t;          // [0, 4096): 256 chunks per row
            int mr = id >> 8;
            int c  = (id & 255) * 2;
            uint32_t dst = ldsA_base + (uint32_t)(((k * 16 + mr) * LDSA_RSTR + c) * 4);
            uint64_t src = (uint64_t)(uintptr_t)(Xp + (size_t)mr * CDIM + c);
            asm volatile("global_load_async_to_lds_b64 %0, %1, off"
                         :: "v"(dst), "v"(src) : "memory");
        }
    }
    asm volatile("s_wait_asynccnt 0" ::: "memory");
    __syncthreads();

    const int lane = t & 31;
    const int hi   = lane >> 4;      // 0: K=0/1 half, 1: K=2/3 half
    const int l16  = lane & 15;      // A row / D column within fragment
    const int n    = t >> 5;         // wave id == batch index

    v8f acc0 = {0,0,0,0,0,0,0,0};
    v8f acc1 = acc0, acc2 = acc0, acc3 = acc0;

    for (int q = 0; q < 40; ++q) {
        int k  = q >> 3;                 // flat = q*4 + j, k = flat/32
        int f0 = (q & 7) << 2;           // f = f0 + j
        int fA = f0 + (hi << 1);         // this lane-half: j = 2*hi (+vgpr)
        int c0 = (fA << 4) + n;          // X column for vgpr0; vgpr1 -> c0+16

        const float* ap = ldsA + (k * 16 + l16) * LDSA_RSTR + c0;
        v2f a;
        a.x = ap[0];
        a.y = ap[16];

        const float* bq = ldsB + (q * 4) * 64 + lane * 2;   // fragment order
        v2f b0 = *(const v2f*)(bq);
        v2f b1 = *(const v2f*)(bq + 64);
        v2f b2 = *(const v2f*)(bq + 128);
        v2f b3 = *(const v2f*)(bq + 192);

        acc0 = __builtin_amdgcn_wmma_f32_16x16x4_f32(false, a, false, b0, (short)0, acc0, false, false);
        acc1 = __builtin_amdgcn_wmma_f32_16x16x4_f32(false, a, false, b1, (short)0, acc1, false, false);
        acc2 = __builtin_amdgcn_wmma_f32_16x16x4_f32(false, a, false, b2, (short)0, acc2, false, false);
        acc3 = __builtin_amdgcn_wmma_f32_16x16x4_f32(false, a, false, b3, (short)0, acc3, false, false);
    }

    // C/D: VGPR v = row v (lanes 0-15) or v+8 (lanes 16-31), col = l16.
    // Pool 4 rows/group: lanes<16 -> pooled rows {0,1}, lanes>=16 -> {2,3}.
    // relu(max(y)+b) == max over relu(y+b): bias is per-column constant.
    const int pbase = blockIdx.x * 4 + (hi << 1);
    v8f accs[4] = {acc0, acc1, acc2, acc3};
#pragma unroll
    for (int tile = 0; tile < 4; ++tile) {
        int   col = tile * 16 + l16;
        float bb  = bias[col];
        v8f   cc  = accs[tile];
        float p0 = fmaxf(fmaxf(cc[0], cc[1]), fmaxf(cc[2], cc[3]));
        float p1 = fmaxf(fmaxf(cc[4], cc[5]), fmaxf(cc[6], cc[7]));
        p0 = fmaxf(p0 + bb, 0.0f);
        p1 = fmaxf(p1 + bb, 0.0f);
        size_t o0 = (((size_t)n * MPOOL + pbase) << 6) + col;
        out[o0]      = p0;
        out[o0 + 64] = p1;      // pooled row pbase+1
    }
}

// ---------------------------------------------------------------------------
extern "C" void kernel_launch(void* const* d_in, const int* in_sizes, int n_in,
                              void* d_out, int out_size, void* d_ws, size_t ws_size,
                              hipStream_t stream) {
    (void)in_sizes; (void)n_in; (void)out_size; (void)ws_size;
    const float* x    = (const float*)d_in[0];
    const int*   rows = (const int*)d_in[1];
    const int*   cols = (const int*)d_in[2];
    const float* vals = (const float*)d_in[3];
    const float* Wk   = (const float*)d_in[4];
    const float* bias = (const float*)d_in[5];
    float*       out  = (float*)d_out;

    float*       Xb  = (float*)d_ws;              // 5 buffers of M*512 f32 (102.4 MB)
    const size_t XSZ = (size_t)MNODES * CDIM;

    int*   cnt    = (int*)(Xb + 5 * XSZ);         // [MNODES]
    int*   offs   = cnt + MNODES;                 // [MNODES+1]
    int*   cursor = offs + MNODES + 1;            // [MNODES]
    int*   scols  = cursor + MNODES;              // [NNZV]
    float* svals  = (float*)(scols + NNZV);       // [NNZV]

    reorder_x<<<625, 512, 0, stream>>>(x, Xb);

    // CSR build (gather SpMM: no float atomics in the hot loops)
    csr_zero   <<<(MNODES + 255) / 256, 256, 0, stream>>>(cnt);
    csr_count  <<<(NNZV + 255) / 256, 256, 0, stream>>>(rows, cnt);
    csr_scan   <<<1, 256, 0, stream>>>(cnt, offs, cursor);
    csr_scatter<<<(NNZV + 255) / 256, 256, 0, stream>>>(rows, cols, vals, cursor, scols, svals);

    for (int step = 1; step < KCH; ++step) {
        const float* Xm1 = Xb + (size_t)(step - 1) * XSZ;
        const float* Xm2 = (step >= 2) ? (Xb + (size_t)(step - 2) * XSZ) : Xm1;
        float*       Y   = Xb + (size_t)step * XSZ;
        float sA = (step == 1) ?  1.0f :  2.0f;
        float a1 = (step == 1) ? -1.0f : -2.0f;
        float a0 = (step == 1) ?  0.0f : -1.0f;
        cheb_spmm_gather<<<1250, 256, 0, stream>>>(offs, scols, svals, Xm1, Xm2, Y, sA, a1, a0);
    }

    cheb_gemm<<<625, 512, (LDSA_TOT + LDSB_TOT) * sizeof(float), stream>>>(Xb, Wk, bias, out);
}